// SoftRoleDecoder_89816356094528
// MI455X (gfx1250) — compile-verified
//
#include <hip/hip_runtime.h>
#include <math.h>

typedef float v2f __attribute__((ext_vector_type(2)));
typedef float v4f __attribute__((ext_vector_type(4)));
typedef float v8f __attribute__((ext_vector_type(8)));

#define Bd 16
#define Sd 512
#define Hd 1024
#define Ed 32
#define Pd 4
#define Rd 8
#define MROWS (Bd * Sd)      // 8192
#define KTOT (2 * Hd)        // 2048
#define NKT (KTOT / 16)      // 128 K-slices

// ---------------------------------------------------------------- utilities
__global__ void zero_kernel(float* __restrict__ p, int n) {
    int stride = gridDim.x * blockDim.x;
    for (int i = blockIdx.x * blockDim.x + threadIdx.x; i < n; i += stride) p[i] = 0.0f;
}

__global__ void masksum_kernel(const float* __restrict__ mask, float* __restrict__ out) {
    __shared__ float red[256];
    float a = 0.0f;
    for (int i = threadIdx.x; i < MROWS; i += 256) a += mask[i];
    red[threadIdx.x] = a;
    __syncthreads();
    for (int o = 128; o > 0; o >>= 1) {
        if (threadIdx.x < o) red[threadIdx.x] += red[threadIdx.x + o];
        __syncthreads();
    }
    if (threadIdx.x == 0) out[0] = red[0];
}

// ------------------------------------------------------- per-row dual GEMV
// one wave32 per (b,s) row; reduces H=1024 with lane-strided loads
__global__ __launch_bounds__(256)
void logits_kernel(const float* __restrict__ sr, const float* __restrict__ tok,
                   const float* __restrict__ ent, const float* __restrict__ pre,
                   const float* __restrict__ sw, const float* __restrict__ sb,
                   const float* __restrict__ mw, const float* __restrict__ mb,
                   float* __restrict__ s_logit, float* __restrict__ m_logit) {
    int row = (blockIdx.x * 256 + threadIdx.x) >> 5;  // 0..8191
    int lane = threadIdx.x & 31;
    const float* srp = sr + (size_t)row * Hd;
    const float* tkp = tok + (size_t)row * Hd;
    const float* enp = ent + (size_t)row * Hd;
    const float* prp = pre + (size_t)row * Hd;
    float ss = 0.0f, ms = 0.0f;
    for (int k = lane; k < Hd; k += 32) {
        float a = srp[k], t = tkp[k], e = enp[k], p = prp[k];
        ss += a * sw[k] + t * sw[Hd + k] + p * sw[2 * Hd + k];
        ms += a * mw[k] + e * mw[Hd + k] + p * mw[2 * Hd + k];
    }
    for (int o = 16; o > 0; o >>= 1) {
        ss += __shfl_xor(ss, o, 32);
        ms += __shfl_xor(ms, o, 32);
    }
    if (lane == 0) {
        s_logit[row] = ss + sb[0];
        m_logit[row] = ms + mb[0];
    }
}

// --------------------------------------- softmax over batch axis (axis=0)
__global__ void colstats_kernel(const float* __restrict__ s_logit,
                                const float* __restrict__ m_logit,
                                float* __restrict__ stats) {
    int s = blockIdx.x * 256 + threadIdx.x;  // 0..511
    float smax = -1e30f, mmax = -1e30f;
    for (int b = 0; b < Bd; ++b) {
        smax = fmaxf(smax, s_logit[b * Sd + s]);
        mmax = fmaxf(mmax, m_logit[b * Sd + s]);
    }
    float ssum = 0.0f, msum = 0.0f;
    for (int b = 0; b < Bd; ++b) {
        ssum += expf(s_logit[b * Sd + s] - smax);
        msum += expf(m_logit[b * Sd + s] - mmax);
    }
    stats[s] = smax; stats[Sd + s] = ssum;
    stats[2 * Sd + s] = mmax; stats[3 * Sd + s] = msum;
}

// scatter-add expressed as deterministic gather (no atomics)
__global__ __launch_bounds__(512)
void scatter_kernel(const float* __restrict__ s_logit, const float* __restrict__ m_logit,
                    const float* __restrict__ stats, const int* __restrict__ c2t,
                    const int* __restrict__ e2t, float* __restrict__ pred_tok,
                    float* __restrict__ escore) {
    __shared__ float sp[Sd], mp[Sd];
    __shared__ int ci[Sd], ei[Sd];
    int b = blockIdx.x, s = threadIdx.x;
    sp[s] = expf(s_logit[b * Sd + s] - stats[s]) / stats[Sd + s];
    mp[s] = expf(m_logit[b * Sd + s] - stats[2 * Sd + s]) / stats[3 * Sd + s];
    ci[s] = c2t[b * Sd + s];
    ei[s] = e2t[b * Sd + s];
    __syncthreads();
    float acc = 0.0f;
    for (int k = 0; k < Sd; ++k)
        if (ci[k] == s) acc += sp[k];      // s>=0 so invalid (-1) never matches
    pred_tok[b * Sd + s] = acc;
    if (s < Ed) {
        float a2 = 0.0f;
        for (int k = 0; k < Sd; ++k)
            if (ei[k] == s) a2 += mp[k];
        escore[b * Ed + s] = a2;
    }
}

// span coverage einsum + max-merge -> d_out slice
__global__ void merge_kernel(const float* __restrict__ pred_tok,
                             const float* __restrict__ escore,
                             const int* __restrict__ spans, float* __restrict__ out_merged) {
    int m = blockIdx.x * 256 + threadIdx.x;  // 0..8191
    int b = m >> 9, s = m & (Sd - 1);
    float pe = 0.0f;
    for (int e = 0; e < Ed; ++e) {
        float sc = escore[b * Ed + e];
        const int* sp = spans + ((b * Ed + e) * Pd) * 2;
#pragma unroll
        for (int p = 0; p < Pd; ++p) {
            int st = sp[p * 2], en = sp[p * 2 + 1];
            pe += (s >= st && s < en) ? sc : 0.0f;
        }
    }
    out_merged[m] = fmaxf(pred_tok[m], pe);
}

// gather merged back to char positions + per-(r,b) NLL
__global__ __launch_bounds__(512)
void arg_nll_kernel(const float* __restrict__ merged, const int* __restrict__ c2t,
                    const int* __restrict__ labels, float* __restrict__ arg,
                    float* __restrict__ nll) {
    __shared__ float row[Sd];
    __shared__ float red[Sd];
    int b = blockIdx.x, s = threadIdx.x;
    float v = merged[b * Sd + s];
    row[s] = v;
    __syncthreads();
    int c = c2t[b * Sd + s];
    arg[b * Sd + s] = (c >= 0) ? row[c] : 0.0f;
    red[s] = v;
    __syncthreads();
    for (int o = 256; o > 0; o >>= 1) {
        if (s < o) red[s] = fmaxf(red[s], red[s + o]);
        __syncthreads();
    }
    float rmax = red[0];
    __syncthreads();
    red[s] = expf(v - rmax);
    __syncthreads();
    for (int o = 256; o > 0; o >>= 1) {
        if (s < o) red[s] += red[s + o];
        __syncthreads();
    }
    if (s == 0) {
        int lab = labels[b];
        nll[b] = -(row[lab] - rmax - logf(red[0]));
    }
}

__global__ void loss_final_kernel(const float* __restrict__ nll, const float* __restrict__ msum,
                                  float* __restrict__ out) {
    __shared__ float red[128];
    int t = threadIdx.x;
    red[t] = nll[t];
    __syncthreads();
    for (int o = 64; o > 0; o >>= 1) {
        if (t < o) red[t] += red[t + o];
        __syncthreads();
    }
    if (t == 0) out[0] = red[0] * msum[0] / (float)Bd;
}

// --------------------------------------------------------- WMMA fp32 GEMM
// new_pre[M=8192, N=1024] = A[M, K=2048] * W[K, N] + bias
// A[m,k] = (k < H) ? token[m,k]*arg[m] : pre[m,k-H]
// Block: 64x128 tile, 256 threads = 8 waves; wave = 16x64 strip (4x v8f acc).
// Software-pipelined: slice kt+1 prefetched to registers during compute of kt.
// Both stage buffers are [row][k] so A and B fragments are single ds_load_b64.
__global__ __launch_bounds__(256)
void gemm_answer_kernel(const float* __restrict__ token, const float* __restrict__ pre,
                        const float* __restrict__ arg, const float* __restrict__ W,
                        const float* __restrict__ bias, float* __restrict__ out) {
    __shared__ float As[64][20];    // [m][k], row stride 80B (16B aligned, bank-skewed)
    __shared__ float Bs[128][20];   // [n][k], row stride 80B

    const int t = threadIdx.x;
    const int bx = blockIdx.x;      // N tile (0..7)
    const int by = blockIdx.y;      // M tile (0..127)
    const int lane = t & 31;
    const int half = lane >> 4;     // 0: lanes 0-15, 1: lanes 16-31
    const int lm = lane & 15;
    const int wave = t >> 5;        // 0..7
    const int rm = (wave >> 1) * 16;   // wave row base within block tile
    const int nb = (wave & 1) * 64;    // wave col base within block tile

    // global->LDS assignments
    const int mA = t >> 2;          // 0..63
    const int kA = (t & 3) * 4;     // 0,4,8,12
    const int mgA = by * 64 + mA;
    const float am = arg[mgA];
    const int nB = t & 127;         // 0..127  (column of W this thread stages)
    const int kB0 = (t >> 7) * 8;   // 0 or 8  (k range within slice)
    const float* wcol = W + (size_t)bx * 128 + nB;

    v8f acc[4] = {};

    // ---- prefetch slice 0 (always in token half)
    v4f av;
    v4f bb0, bb1;
    {
        av = *(const v4f*)(token + (size_t)mgA * Hd + kA);
        av = av * am;
#pragma unroll
        for (int j = 0; j < 4; ++j) bb0[j] = wcol[(size_t)(kB0 + j) * Hd];
#pragma unroll
        for (int j = 0; j < 4; ++j) bb1[j] = wcol[(size_t)(kB0 + 4 + j) * Hd];
    }

    for (int kt = 0; kt < NKT; ++kt) {
        __syncthreads();                       // previous compute done with LDS
        *(v4f*)&As[mA][kA] = av;
        *(v4f*)&Bs[nB][kB0] = bb0;
        *(v4f*)&Bs[nB][kB0 + 4] = bb1;
        __syncthreads();                       // LDS slice ready

        // ---- prefetch slice kt+1 into registers (overlaps with WMMA below)
        if (kt + 1 < NKT) {
            int kg = (kt + 1) * 16 + kA;
            if (kg < Hd) {                     // uniform branch (16 | 1024)
                av = *(const v4f*)(token + (size_t)mgA * Hd + kg);
                av = av * am;
            } else {
                av = *(const v4f*)(pre + (size_t)mgA * Hd + (kg - Hd));
            }
            int kgB = (kt + 1) * 16 + kB0;
#pragma unroll
            for (int j = 0; j < 4; ++j) bb0[j] = wcol[(size_t)(kgB + j) * Hd];
#pragma unroll
            for (int j = 0; j < 4; ++j) bb1[j] = wcol[(size_t)(kgB + 4 + j) * Hd];
        }

        // ---- 16 WMMAs on the staged slice
#pragma unroll
        for (int kk = 0; kk < 16; kk += 4) {
            // A 16x4 frag: lane=M row; v[0]=K(kk+2*half), v[1]=K(kk+2*half+1)
            v2f a = *(const v2f*)&As[rm + lm][kk + 2 * half];
#pragma unroll
            for (int n = 0; n < 4; ++n) {
                // B 4x16 frag: lane=N col; v[0]=K(kk+2*half), v[1]=K(kk+2*half+1)
                v2f b = *(const v2f*)&Bs[nb + n * 16 + lm][kk + 2 * half];
                acc[n] = __builtin_amdgcn_wmma_f32_16x16x4_f32(
                    false, a, false, b, (short)0, acc[n], false, false);
            }
        }
    }

    // epilogue: C layout -> VGPR j: M = rm + j + 8*half, N = lane&15
#pragma unroll
    for (int n = 0; n < 4; ++n) {
        int ng = bx * 128 + nb + n * 16 + lm;
        float bsv = bias[ng];
#pragma unroll
        for (int j = 0; j < 8; ++j) {
            int mg = by * 64 + rm + j + half * 8;
            out[(size_t)mg * Hd + ng] = acc[n][j] + bsv;
        }
    }
}

// ---------------------------------------------------------------- launcher
extern "C" void kernel_launch(void* const* d_in, const int* in_sizes, int n_in,
                              void* d_out, int out_size, void* d_ws, size_t ws_size,
                              hipStream_t stream) {
    const int*   labels = (const int*)d_in[0];     // (R,B)
    const float* summar = (const float*)d_in[1];   // (R,B,S,H)
    const float* token  = (const float*)d_in[2];   // (B,S,H)
    const float* ents   = (const float*)d_in[3];   // (B,S,H)
    const float* tmask  = (const float*)d_in[4];   // (B,S)
    // d_in[5] entity_mask: unused by reference
    const int*   spans  = (const int*)d_in[6];     // (B,E,P,2)
    const int*   c2t    = (const int*)d_in[7];     // (B,S)
    const int*   e2t    = (const int*)d_in[8];     // (B,S)
    const float* sw = (const float*)d_in[9];
    const float* sb = (const float*)d_in[10];
    const float* mw = (const float*)d_in[11];
    const float* mb = (const float*)d_in[12];
    const float* aw = (const float*)d_in[13];      // (2H,H)
    const float* ab = (const float*)d_in[14];      // (H,)

    float* out = (float*)d_out;                    // [0]=loss, [1..]=pred_arg_ids (R,B,S)
    float* ws = (float*)d_ws;

    const size_t PRE_N = (size_t)MROWS * Hd;       // 8,388,608 floats
    float* pre0  = ws;
    float* pre1  = pre0 + PRE_N;
    float* slog  = pre1 + PRE_N;          // 8192
    float* mlog  = slog + MROWS;          // 8192
    float* stats = mlog + MROWS;          // 2048
    float* ptok  = stats + 4 * Sd;        // 8192
    float* esc   = ptok + MROWS;          // 512
    float* argb  = esc + Bd * Ed;         // 8192
    float* nll   = argb + MROWS;          // 128
    float* msum  = nll + Rd * Bd;         // 1

    zero_kernel<<<2048, 256, 0, stream>>>(pre0, (int)PRE_N);
    masksum_kernel<<<1, 256, 0, stream>>>(tmask, msum);

    float* pre = pre0;
    float* nxt = pre1;
    for (int r = 0; r < Rd; ++r) {
        const float* sr = summar + (size_t)r * PRE_N;
        logits_kernel<<<1024, 256, 0, stream>>>(sr, token, ents, pre, sw, sb, mw, mb,
                                                slog, mlog);
        colstats_kernel<<<2, 256, 0, stream>>>(slog, mlog, stats);
        scatter_kernel<<<Bd, 512, 0, stream>>>(slog, mlog, stats, c2t, e2t, ptok, esc);
        float* merged = out + 1 + (size_t)r * MROWS;
        merge_kernel<<<32, 256, 0, stream>>>(ptok, esc, spans, merged);
        arg_nll_kernel<<<Bd, 512, 0, stream>>>(merged, c2t, labels + r * Bd, argb,
                                               nll + r * Bd);
        dim3 g(Hd / 128, MROWS / 64);
        gemm_answer_kernel<<<g, 256, 0, stream>>>(token, pre, argb, aw, ab, nxt);
        float* tp = pre; pre = nxt; nxt = tp;
    }
    loss_final_kernel<<<1, 128, 0, stream>>>(nll, msum, out);
}